// Channel_M_16965120819475
// MI455X (gfx1250) — compile-verified
//
#include <hip/hip_runtime.h>
#include <stdint.h>

// DNA channel simulator (sub -> del -> ins), 4 realizations per row.
// grid.x = B rows; block = 128 threads = 4 wave32s; wave c simulates channel
// realization c of the row. The input row is staged once into LDS with the
// CDNA5 async global->LDS DMA and shared by all 4 waves.
//
// Memory-bound: 33.5 MB in + 134.7 MB out => ~7.2 us floor at 23.3 TB/s.
// Output is streamed with non-temporal b64 stores (rows are 8B- but not
// 16B-aligned: 514 floats/row), input rows ride the async-LDS path.

#define LSEQ 512
#define LOUT 514            // L + 2
#define NCH 4
#define SOUT_STRIDE 520     // floats; 2080 B row stride (16B multiple)

typedef float v2f __attribute__((ext_vector_type(2)));   // native vector for NT builtins
typedef float v4f __attribute__((ext_vector_type(4)));

__device__ __forceinline__ uint32_t mix32(uint32_t x) {
  x ^= x >> 16; x *= 0x7feb352du;
  x ^= x >> 15; x *= 0x846ca68bu;
  x ^= x >> 16;
  return x;
}

// Wave32 exclusive scan; also returns the wave total.
__device__ __forceinline__ int wave_excl_scan_total(int v, int lane, int& total) {
  int incl = v;
  #pragma unroll
  for (int d = 1; d < 32; d <<= 1) {
    int t = __shfl_up(incl, d, 32);
    if (lane >= d) incl += t;
  }
  total = __shfl(incl, 31, 32);
  return incl - v;
}

__global__ __launch_bounds__(128)
void channel_m_kernel(const float* __restrict__ seg,
                      const float* __restrict__ sub_p,
                      const float* __restrict__ del_p,
                      const float* __restrict__ ins_p,
                      float* __restrict__ out) {
  __shared__ __align__(16) float s_in[LSEQ];
  __shared__ __align__(16) float s_comp[NCH][LSEQ];
  __shared__ __align__(16) float s_out[NCH][SOUT_STRIDE];

  const int b    = blockIdx.x;
  const int tid  = threadIdx.x;
  const int c    = tid >> 5;    // wave id == channel realization
  const int lane = tid & 31;

  // ---- Async-stage the input row into LDS (CDNA5 async global->LDS DMA).
  // 128 lanes x 16B = 2048B = full 512-float row. GVS addressing:
  // global = SADDR(sgpr pair) + VADDR(u32 offset); VDST holds LDS byte addr.
  {
    const uint64_t gbase = (uint64_t)(uintptr_t)(seg + (size_t)b * LSEQ);
    const uint32_t voff  = (uint32_t)tid * 16u;
    const uint32_t ldsa  = (uint32_t)(uintptr_t)(&s_in[0]) + (uint32_t)tid * 16u;
    asm volatile("global_load_async_to_lds_b128 %0, %1, %2"
                 :: "v"(ldsa), "v"(voff), "s"(gbase)
                 : "memory");
    asm volatile("s_wait_asynccnt 0" ::: "memory");
  }
  __syncthreads();

  // Error rates -> integer thresholds (15-bit for fused sub/del draws,
  // 24-bit for the insertion draw).
  const uint32_t t_sub = (uint32_t)(sub_p[0] * 32768.0f);
  const uint32_t t_del = (uint32_t)(del_p[0] * 32768.0f);
  const uint32_t t_ins = (uint32_t)(ins_p[0] * 16777216.0f);

  // Stateless counter-based RNG seeds per (row, channel, stage).
  const uint32_t rowc    = ((uint32_t)b << 2) | (uint32_t)c;
  const uint32_t seed_sd = mix32(rowc ^ 0x85ebca6bu);
  const uint32_t seed_in = mix32(rowc ^ 0x27d4eb2fu);

  // ---- Vectorized LDS read: lane owns 16 contiguous symbols (4x b128) ----
  const int p0 = lane * 16;
  float vals[16];
  {
    const v4f* s_in4 = (const v4f*)s_in;
    #pragma unroll
    for (int k = 0; k < 4; ++k) {
      const v4f t = s_in4[lane * 4 + k];
      vals[4 * k + 0] = t.x; vals[4 * k + 1] = t.y;
      vals[4 * k + 2] = t.z; vals[4 * k + 3] = t.w;
    }
  }

  // ---- Substitution + deletion flags: ONE hash per element serves the
  // 2-bit substitution symbol, the sub Bernoulli, and the keep flag ----
  uint32_t keepm = 0;
  int kcnt = 0;
  #pragma unroll
  for (int j = 0; j < 16; ++j) {
    const int p = p0 + j;
    const uint32_t h = mix32(seed_sd + (uint32_t)p * 0x9e3779b9u);
    int xi = (int)vals[j];                                  // symbol in {0..3}
    const int sub_on = (((h >> 2) & 0x7fffu) < t_sub);
    xi = (xi + (int)(h & 3u) * sub_on) & 3;                 // (x + d*mask) mod 4
    vals[j] = (float)xi;
    const int k = ((h >> 17) >= t_del);                     // keep
    keepm |= (uint32_t)k << j;
    kcnt  += k;
  }

  int len;                                                  // survivors (<=512)
  const int kbase = wave_excl_scan_total(kcnt, lane, len);

  // Stream-compact survivors into the wave-private LDS buffer.
  {
    int widx = kbase;
    #pragma unroll
    for (int j = 0; j < 16; ++j)
      if (keepm & (1u << j)) s_comp[c][widx++] = vals[j];
  }

  // Zero the output row (background (-1) + 1.0 == 0.0): 4x ds_store_b128
  // across the wave covers 512 floats; lane 0 zeroes the 2-float tail.
  {
    v4f* o4 = (v4f*)&s_out[c][0];                           // 16B-aligned row
    const v4f z4 = (v4f)(0.0f);
    #pragma unroll
    for (int j = 0; j < 4; ++j) o4[j * 32 + lane] = z4;
    if (lane == 0) ((v2f*)&s_out[c][0])[256] = (v2f)(0.0f);
  }
  // Wave-private LDS ordering fence (DS ops are in-order per wave; the asm
  // memory clobber pins compiler ordering, the wait covers VGPR hazards).
  asm volatile("s_wait_dscnt 0" ::: "memory");

  // ---- Insertion over the compacted sequence (indices i < len only; the
  // reference's trailing -1 padding writes equal the -1 background) ----
  uint32_t insm = 0, syms = 0;
  int icnt = 0;
  #pragma unroll
  for (int j = 0; j < 16; ++j) {
    const int i = p0 + j;
    const uint32_t h = mix32(seed_in + (uint32_t)i * 0x9e3779b9u);
    const int fl = (i < len) && ((h >> 8) < t_ins);
    insm |= (uint32_t)fl << j;
    syms |= (h & 3u) << (2 * j);
    icnt += fl;
  }
  int itotal;
  const int ebase = wave_excl_scan_total(icnt, lane, itotal);

  {
    int e = ebase;                                          // inserts strictly before i
    #pragma unroll
    for (int j = 0; j < 16; ++j) {
      const int i = p0 + j;
      if (i < len) {
        const int slot = i + e;                             // pos_orig
        if (slot < LOUT) s_out[c][slot] = s_comp[c][i] + 1.0f;
        if (insm & (1u << j)) {
          if (slot + 1 < LOUT)                              // pos_ins (never collides)
            s_out[c][slot + 1] = (float)((syms >> (2 * j)) & 3u) + 1.0f;
          ++e;
        }
      }
    }
  }
  asm volatile("s_wait_dscnt 0" ::: "memory");

  // ---- Coalesced non-temporal b64 writeout: out[b][c][0..513].
  // Row byte offset is 8-mod-16, so float2 is the widest safe vector. ----
  {
    v2f* orow2 = (v2f*)(out + ((size_t)b * NCH + (size_t)c) * LOUT);
    const v2f* s2 = (const v2f*)&s_out[c][0];
    #pragma unroll
    for (int j = 0; j < 8; ++j)
      __builtin_nontemporal_store(s2[j * 32 + lane], orow2 + j * 32 + lane);
    if (lane == 0)
      __builtin_nontemporal_store(s2[256], orow2 + 256);    // floats 512..513
  }
}

extern "C" void kernel_launch(void* const* d_in, const int* in_sizes, int n_in,
                              void* d_out, int out_size, void* d_ws, size_t ws_size,
                              hipStream_t stream) {
  const float* seg   = (const float*)d_in[0];
  const float* sub_e = (const float*)d_in[1];
  const float* del_e = (const float*)d_in[2];
  const float* ins_e = (const float*)d_in[3];
  float* out = (float*)d_out;

  const int B = in_sizes[0] / LSEQ;   // 16384
  channel_m_kernel<<<B, 128, 0, stream>>>(seg, sub_e, del_e, ins_e, out);
}